// GCN_78331613544889
// MI455X (gfx1250) — compile-verified
//
#include <hip/hip_runtime.h>
#include <hip/hip_bf16.h>

typedef float v2f __attribute__((ext_vector_type(2)));
typedef float v8f __attribute__((ext_vector_type(8)));

#define N_NODES 100000
#define N_EDGES 1600000
#define FEAT    64

// ---------------------------------------------------------------------------
// C[M x 64] = A[M x K] @ W[K x 64] using V_WMMA_F32_16X16X4_F32.
// One wave computes one 16x16 tile; 4 waves per block cover N=64.
// grid.x = M/16 (M is a multiple of 16), block = 128 threads (4 waves).
//
// VGPR layouts (CDNA5 ISA 7.12.2, 32-bit data, wave32):
//   A (16x4):  lanes 0-15: row M=lane,   vgpr{0,1} = K{0,1}
//              lanes16-31: row M=lane-16,vgpr{0,1} = K{2,3}
//   B (4x16):  lanes 0-15: col N=lane,   vgpr{0,1} = K{0,1}
//              lanes16-31: col N=lane-16,vgpr{0,1} = K{2,3}
//   D (16x16): vgpr r: lanes 0-15 -> row m0+r, lanes 16-31 -> row m0+8+r,
//              col = n0 + (lane & 15)
// ---------------------------------------------------------------------------
__global__ void gemm_wmma(const float* __restrict__ A,
                          const float* __restrict__ W,
                          float* __restrict__ C,
                          int K) {
  const int wave = threadIdx.x >> 5;
  const int lane = threadIdx.x & 31;
  const int half = lane >> 4;        // 0: lanes 0-15, 1: lanes 16-31
  const int l    = lane & 15;
  const int m0   = blockIdx.x * 16;
  const int n0   = wave * 16;

  const float* __restrict__ arow = A + (size_t)(m0 + l) * K;

  v8f acc = {};
  for (int k = 0; k < K; k += 4) {
    const int ka = k + half * 2;
    v2f a, b;
    a.x = arow[ka];
    a.y = arow[ka + 1];
    b.x = W[(size_t)ka * FEAT + n0 + l];
    b.y = W[(size_t)(ka + 1) * FEAT + n0 + l];
    acc = __builtin_amdgcn_wmma_f32_16x16x4_f32(
        /*neg_a=*/false, a, /*neg_b=*/false, b,
        /*c_mod=*/(short)0, acc, /*reuse_a=*/false, /*reuse_b=*/false);
  }

  float* __restrict__ crow = C + (size_t)(m0 + half * 8) * FEAT + n0 + l;
#pragma unroll
  for (int r = 0; r < 8; ++r) {
    crow[(size_t)r * FEAT] = acc[r];
  }
}

// out[n*64 + c] = bias[c]  (folds the +b into the aggregate-buffer init)
__global__ void init_bias(float* __restrict__ out,
                          const float* __restrict__ bias,
                          int total) {
  int i = blockIdx.x * blockDim.x + threadIdx.x;
  if (i < total) out[i] = bias[i & (FEAT - 1)];
}

// For each edge e: out[dst[e], :] += hw[src[e], :]
// 16 threads per edge, each handling 4 channels via float4 load + 4 atomics.
__global__ void scatter_add(const float* __restrict__ hw,
                            const int* __restrict__ src,
                            const int* __restrict__ dst,
                            float* __restrict__ out) {
  int t = blockIdx.x * blockDim.x + threadIdx.x;
  int e = t >> 4;
  if (e >= N_EDGES) return;
  int c = (t & 15) * 4;
  int s = src[e];
  int d = dst[e];
  const float4 v = *reinterpret_cast<const float4*>(hw + (size_t)s * FEAT + c);
  float* o = out + (size_t)d * FEAT + c;
  atomicAdd(o + 0, v.x);
  atomicAdd(o + 1, v.y);
  atomicAdd(o + 2, v.z);
  atomicAdd(o + 3, v.w);
}

__global__ void relu_k(float* __restrict__ x, int total) {
  int i = blockIdx.x * blockDim.x + threadIdx.x;
  if (i < total) x[i] = fmaxf(x[i], 0.0f);
}

extern "C" void kernel_launch(void* const* d_in, const int* in_sizes, int n_in,
                              void* d_out, int out_size, void* d_ws, size_t ws_size,
                              hipStream_t stream) {
  const float* x        = (const float*)d_in[0];
  const int*   edge_src = (const int*)d_in[1];
  const int*   edge_dst = (const int*)d_in[2];
  const float* W1       = (const float*)d_in[3];
  const float* b1       = (const float*)d_in[4];
  const float* W2       = (const float*)d_in[5];
  const float* b2       = (const float*)d_in[6];
  const float* W3       = (const float*)d_in[7];
  const float* b3       = (const float*)d_in[8];
  float* out = (float*)d_out;

  const int feat_total = N_NODES * FEAT;            // 6.4M floats
  float* hw = (float*)d_ws;                         // [N_NODES x 64]
  float* h  = hw + (size_t)N_NODES * FEAT;          // [N_NODES x 64]

  const dim3 gemm_grid(N_NODES / 16);
  const dim3 gemm_blk(128);
  const int elt_blocks  = (feat_total + 255) / 256;
  const int scat_blocks = (N_EDGES * 16 + 255) / 256;

  // ---- layer 1: h = relu(scatter(x @ W1) + b1) ----
  gemm_wmma<<<gemm_grid, gemm_blk, 0, stream>>>(x, W1, hw, 128);
  init_bias<<<elt_blocks, 256, 0, stream>>>(h, b1, feat_total);
  scatter_add<<<scat_blocks, 256, 0, stream>>>(hw, edge_src, edge_dst, h);
  relu_k<<<elt_blocks, 256, 0, stream>>>(h, feat_total);

  // ---- layer 2: h = relu(scatter(h @ W2) + b2) ----
  gemm_wmma<<<gemm_grid, gemm_blk, 0, stream>>>(h, W2, hw, 64);
  init_bias<<<elt_blocks, 256, 0, stream>>>(h, b2, feat_total);
  scatter_add<<<scat_blocks, 256, 0, stream>>>(hw, edge_src, edge_dst, h);
  relu_k<<<elt_blocks, 256, 0, stream>>>(h, feat_total);

  // ---- layer 3: out = scatter(h @ W3) + b3 ----
  gemm_wmma<<<gemm_grid, gemm_blk, 0, stream>>>(h, W3, hw, 64);
  init_bias<<<elt_blocks, 256, 0, stream>>>(out, b3, feat_total);
  scatter_add<<<scat_blocks, 256, 0, stream>>>(hw, edge_src, edge_dst, out);
}